// PatternBasedV2_260
// MI455X (gfx1250) — compile-verified
//
#include <hip/hip_runtime.h>
#include <hip/hip_bf16.h>

typedef float v2f __attribute__((ext_vector_type(2)));
typedef float v8f __attribute__((ext_vector_type(8)));

#define FRONT 256
#define MID   64
#define BACK  32
#define NPAT  16

__constant__ unsigned long long PAT_MASK[NPAT] = {
    16639ull, 65280ull, 16711680ull, 4278190080ull, 3871ull, 198415ull,
    4345695256ull, 1108169199648ull, 283691315109952ull, 72624976668147840ull,
    7357ull, 135137027ull, 460551ull, 1279ull, 134614787ull, 33693443ull
};

// ---------------------------------------------------------------------------
// Kernel A: per-board index computation + embedding-bag gather with max_norm.
// One block (256 threads = 8 wave32) per board.
// ---------------------------------------------------------------------------
__global__ __launch_bounds__(256) void gather_bag_kernel(
    const int* __restrict__ x,        // [B,2,8,8]
    const float* __restrict__ emb,    // [225990, 256]
    float* __restrict__ m_out)        // [B, 256]
{
    __shared__ int   v[64];
    __shared__ int   sIdx[128];
    __shared__ float mAcc[FRONT];

    const int b = blockIdx.x;
    const int t = threadIdx.x;

    mAcc[t] = 0.0f;
    if (t < 64) {
        int p0 = x[b * 128 + t];
        int p1 = x[b * 128 + 64 + t];
        v[t] = p0 + 2 * p1;                 // cell in {0,1,2}
    }
    __syncthreads();

    if (t < 128) {
        const int sym = t >> 4;
        const int pat = t & 15;
        // per-pattern offset into the global table: sum of 3^popcount(prev)
        long long bias = 0;
        for (int q = 0; q < pat; ++q) {
            int n = __popcll(PAT_MASK[q]);
            long long p = 1;
            for (int i = 0; i < n; ++i) p *= 3;
            bias += p;
        }
        const unsigned long long mask = PAT_MASK[pat];
        long long idx = bias;
        long long p3  = 1;
        for (int pos = 0; pos < 64; ++pos) {
            if ((mask >> pos) & 1ull) {
                int r  = pos >> 3, c = pos & 7;
                int c2 = (sym & 4) ? (7 - c) : c;     // col flip
                int r2 = (sym & 2) ? (7 - r) : r;     // row flip
                int src = (sym & 1) ? (c2 * 8 + r2)   // transpose
                                    : (r2 * 8 + c2);
                idx += (long long)v[src] * p3;
                p3 *= 3;
            }
        }
        sIdx[t] = (int)idx;
    }
    __syncthreads();

    // 8 waves x 16 rows each; lane covers 8 of 256 cols (two float4 chunks)
    const int lane = t & 31;
    const int wave = t >> 5;
    float4 acc0 = make_float4(0.f, 0.f, 0.f, 0.f);
    float4 acc1 = make_float4(0.f, 0.f, 0.f, 0.f);

    for (int j = wave * 16; j < wave * 16 + 16; ++j) {
        const long long idx = sIdx[j];
        const float* row = emb + idx * (long long)FRONT;
        float4 r0 = *(const float4*)(row + 4 * lane);          // cols 0..127
        float4 r1 = *(const float4*)(row + 128 + 4 * lane);    // cols 128..255
        float ss = r0.x * r0.x + r0.y * r0.y + r0.z * r0.z + r0.w * r0.w
                 + r1.x * r1.x + r1.y * r1.y + r1.z * r1.z + r1.w * r1.w;
        #pragma unroll
        for (int off = 16; off > 0; off >>= 1)
            ss += __shfl_xor(ss, off);                         // wave32 reduce
        float norm  = sqrtf(ss);
        float scale = (norm > 1.0f) ? (1.0f / (norm + 1e-7f)) : 1.0f;
        acc0.x += r0.x * scale; acc0.y += r0.y * scale;
        acc0.z += r0.z * scale; acc0.w += r0.w * scale;
        acc1.x += r1.x * scale; acc1.y += r1.y * scale;
        acc1.z += r1.z * scale; acc1.w += r1.w * scale;
    }
    atomicAdd(&mAcc[4 * lane + 0], acc0.x);
    atomicAdd(&mAcc[4 * lane + 1], acc0.y);
    atomicAdd(&mAcc[4 * lane + 2], acc0.z);
    atomicAdd(&mAcc[4 * lane + 3], acc0.w);
    atomicAdd(&mAcc[128 + 4 * lane + 0], acc1.x);
    atomicAdd(&mAcc[128 + 4 * lane + 1], acc1.y);
    atomicAdd(&mAcc[128 + 4 * lane + 2], acc1.z);
    atomicAdd(&mAcc[128 + 4 * lane + 3], acc1.w);
    __syncthreads();

    m_out[(long long)b * FRONT + t] = mAcc[t];
}

// ---------------------------------------------------------------------------
// Kernel B: fused 3-layer MLP using V_WMMA_F32_16X16X4_F32 (fp32 WMMA).
// Each wave32 owns a 16-board tile. 8 waves per block.
// A frag (16x4 f32): lanes 0-15 hold K=0,1 ; lanes 16-31 hold K=2,3.
// C/D (16x16 f32):   VGPR j = row j (lanes 0-15) / row j+8 (lanes 16-31).
// ---------------------------------------------------------------------------
__global__ __launch_bounds__(256) void mlp_wmma_kernel(
    const float* __restrict__ m,   // [B, 256]
    const float* __restrict__ w1,  // [256, 64]
    const float* __restrict__ b1,  // [64]
    const float* __restrict__ w2,  // [64, 32]
    const float* __restrict__ b2,  // [32]
    const float* __restrict__ w3,  // [32, 1]
    const float* __restrict__ b3,  // [1]
    float* __restrict__ out)       // [B]
{
    __shared__ float h1[8][16][MID];
    __shared__ float h2[8][16][BACK];

    const int lane    = threadIdx.x & 31;
    const int wave    = threadIdx.x >> 5;
    const int tile    = blockIdx.x * 8 + wave;
    const int rowBase = tile * 16;
    const int halfK   = (lane >> 4) << 1;   // 0 or 2
    const int nlane   = lane & 15;          // M for A-frag, N for B-frag
    const int mrow    = (lane >> 4) << 3;   // 0 or 8 (C/D row base)

    // ---- layer 1: [16x256] @ [256x64] ----
    v8f c1[4] = {};
    for (int k0 = 0; k0 < FRONT; k0 += 4) {
        const int aK = k0 + halfK;
        v2f a = *(const v2f*)(m + (long long)(rowBase + nlane) * FRONT + aK);
        #pragma unroll
        for (int n = 0; n < 4; ++n) {
            v2f bb;
            bb.x = w1[aK * MID + n * 16 + nlane];
            bb.y = w1[(aK + 1) * MID + n * 16 + nlane];
            c1[n] = __builtin_amdgcn_wmma_f32_16x16x4_f32(
                false, a, false, bb, (short)0, c1[n], false, false);
        }
    }
    #pragma unroll
    for (int n = 0; n < 4; ++n) {
        float bias = b1[n * 16 + nlane];
        #pragma unroll
        for (int j = 0; j < 8; ++j)
            h1[wave][mrow + j][n * 16 + nlane] = fmaxf(c1[n][j] + bias, 0.0f);
    }
    __syncthreads();

    // ---- layer 2: [16x64] @ [64x32] ----
    v8f c2[2] = {};
    for (int k0 = 0; k0 < MID; k0 += 4) {
        const int aK = k0 + halfK;
        v2f a;
        a.x = h1[wave][nlane][aK];
        a.y = h1[wave][nlane][aK + 1];
        #pragma unroll
        for (int n = 0; n < 2; ++n) {
            v2f bb;
            bb.x = w2[aK * BACK + n * 16 + nlane];
            bb.y = w2[(aK + 1) * BACK + n * 16 + nlane];
            c2[n] = __builtin_amdgcn_wmma_f32_16x16x4_f32(
                false, a, false, bb, (short)0, c2[n], false, false);
        }
    }
    #pragma unroll
    for (int n = 0; n < 2; ++n) {
        float bias = b2[n * 16 + nlane];
        #pragma unroll
        for (int j = 0; j < 8; ++j)
            h2[wave][mrow + j][n * 16 + nlane] = fmaxf(c2[n][j] + bias, 0.0f);
    }
    __syncthreads();

    // ---- layer 3: [16x32] @ [32x1] ----
    if (lane < 16) {
        float s = b3[0];
        #pragma unroll
        for (int k = 0; k < BACK; ++k)
            s += h2[wave][lane][k] * w3[k];
        out[rowBase + lane] = s;
    }
}

extern "C" void kernel_launch(void* const* d_in, const int* in_sizes, int n_in,
                              void* d_out, int out_size, void* d_ws, size_t ws_size,
                              hipStream_t stream) {
    const int*   x   = (const int*)d_in[0];
    const float* emb = (const float*)d_in[1];
    const float* w1  = (const float*)d_in[2];
    const float* b1  = (const float*)d_in[3];
    const float* w2  = (const float*)d_in[4];
    const float* b2  = (const float*)d_in[5];
    const float* w3  = (const float*)d_in[6];
    const float* b3  = (const float*)d_in[7];
    float* out = (float*)d_out;

    const int B = in_sizes[0] / 128;          // [B,2,8,8] -> B
    float* m_ws = (float*)d_ws;               // B*256 floats (8 MB for B=8192)

    gather_bag_kernel<<<B, 256, 0, stream>>>(x, emb, m_ws);
    mlp_wmma_kernel<<<B / 128, 256, 0, stream>>>(m_ws, w1, b1, w2, b2, w3, b3, out);
}